// TanhAttention_79456894976529
// MI455X (gfx1250) — compile-verified
//
#include <hip/hip_runtime.h>
#include <hip/hip_bf16.h>

// Problem sizes (fixed by the reference)
#define B_  4
#define LX  512
#define LM  512
#define DD  256

typedef __attribute__((ext_vector_type(2))) float v2f;
typedef __attribute__((ext_vector_type(8))) float v8f;

__device__ __forceinline__ float fast_tanh(float x) {
#if __has_builtin(__builtin_amdgcn_tanhf)
    return __builtin_amdgcn_tanhf(x);
#else
    float r;
    asm volatile("v_tanh_f32 %0, %1" : "=v"(r) : "v"(x));
    return r;
#endif
}

// ---------------------------------------------------------------------------
// Kernel 1: Y[r, e] = sum_d X[r, d] * W[e, d]  (+ bias[e])
// One wave per 16x16 output tile, K-loop of V_WMMA_F32_16X16X4_F32.
//
// f32 WMMA operand layouts (CDNA5 ISA 7.12.2):
//   A 16x4 : lane<16 -> {v0,v1} = A[M=lane, K=0,1]; lane>=16 -> K=2,3
//   B 4x16 : lane<16 -> {v0,v1} = B[K=0,1, N=lane]; lane>=16 -> K=2,3
//   C/D    : VGPR v -> D[M=v, N=lane]       (lanes 0..15)
//                   -> D[M=8+v, N=lane-16]  (lanes 16..31)
// Since B[k, n] = W[n, k], both fragments load as row-major 2-float chunks.
// ---------------------------------------------------------------------------
__global__ void __launch_bounds__(256)
gemm16x16_wmma_f32(const float* __restrict__ X,    // rows x DD
                   const float* __restrict__ W,    // DD x DD (row-major, Y = X*W^T)
                   const float* __restrict__ bias, // DD or nullptr
                   float*       __restrict__ Y,    // rows x DD
                   int row_tiles)
{
    const int wid  = (blockIdx.x * blockDim.x + threadIdx.x) >> 5;
    const int lane = threadIdx.x & 31;
    const int tile_e = wid & (DD / 16 - 1);   // 16 column tiles
    const int tile_r = wid >> 4;
    if (tile_r >= row_tiles) return;

    const int half = lane >> 4;
    const int l15  = lane & 15;

    const float* arow = X + (size_t)(tile_r * 16 + l15) * DD + 2 * half;
    const float* brow = W + (size_t)(tile_e * 16 + l15) * DD + 2 * half;

    v8f c = {};
#pragma unroll 4
    for (int k = 0; k < DD; k += 4) {
        v2f a = *(const v2f*)(arow + k);
        v2f b = *(const v2f*)(brow + k);
        // 8 args: (neg_a, A, neg_b, B, c_mod, C, reuse_a, reuse_b)
        c = __builtin_amdgcn_wmma_f32_16x16x4_f32(false, a, false, b,
                                                  (short)0, c, false, false);
    }

    const int   col = tile_e * 16 + l15;
    const float bs  = bias ? bias[col] : 0.0f;
    float* yout = Y + (size_t)(tile_r * 16 + half * 8) * DD + col;
#pragma unroll
    for (int v = 0; v < 8; ++v)
        yout[(size_t)v * DD] = c[v] + bs;
}

// ---------------------------------------------------------------------------
// Kernel 2: S[b,x,m] = sum_d tanh(item1[b,x,d] + item2[b,m,d]) * Wt[d]
// Block = 32x8 threads -> 8 x-rows by 32 m-cols output tile.
// item2 tile stored transposed in LDS with stride 33 (bank-conflict free).
// item1 reads are wave-uniform broadcasts.
// ---------------------------------------------------------------------------
__global__ void __launch_bounds__(256)
tanh_attn_kernel(const float* __restrict__ item1, // (B*LX) x DD
                 const float* __restrict__ item2, // (B*LM) x DD
                 const int*   __restrict__ mask,  // B x LM
                 const float* __restrict__ Wt,    // DD
                 float*       __restrict__ out)   // B x LX x LM
{
    __shared__ float s1[8][DD];       // 8 rows of item1
    __shared__ float s2[DD][33];      // item2 tile, transposed + padded
    __shared__ float swt[DD];

    const int tx = threadIdx.x;       // 0..31 : m within tile
    const int ty = threadIdx.y;       // 0..7  : x within tile
    const int t  = ty * 32 + tx;      // 0..255
    const int b  = blockIdx.z;
    const int x0 = blockIdx.y * 8;
    const int m0 = blockIdx.x * 32;

    // item1 tile: 8 rows x 256 cols, thread t loads column t of each row
#pragma unroll
    for (int i = 0; i < 8; ++i)
        s1[i][t] = item1[(size_t)(b * LX + x0 + i) * DD + t];

    // item2 tile: 32 rows x 256 cols, stored transposed s2[d][m]
#pragma unroll
    for (int j = 0; j < 32; ++j)
        s2[t][j] = item2[(size_t)(b * LM + m0 + j) * DD + t];

    swt[t] = Wt[t];
    __syncthreads();

    float acc = 0.0f;
#pragma unroll 4
    for (int d = 0; d < DD; ++d)
        acc += fast_tanh(s1[ty][d] + s2[d][tx]) * swt[d];

    const int mk = mask[b * LM + m0 + tx];
    out[(size_t)(b * LX + x0 + ty) * LM + m0 + tx] = (mk == 0) ? -10000.0f : acc;
}

// ---------------------------------------------------------------------------
extern "C" void kernel_launch(void* const* d_in, const int* in_sizes, int n_in,
                              void* d_out, int out_size, void* d_ws, size_t ws_size,
                              hipStream_t stream) {
    const float* x    = (const float*)d_in[0]; // (B, LX, D)
    const float* mem  = (const float*)d_in[1]; // (B, LM, D)
    const int*   mask = (const int*)d_in[2];   // (B, LM)
    const float* W1   = (const float*)d_in[3]; // (D, D)
    const float* b1   = (const float*)d_in[4]; // (D)
    const float* W2   = (const float*)d_in[5]; // (D, D)
    const float* Wt   = (const float*)d_in[6]; // (D)
    float* out = (float*)d_out;                // (B, LX, LM)

    float* item1 = (float*)d_ws;                       // 2 MB
    float* item2 = item1 + (size_t)B_ * LX * DD;       // 2 MB

    // item1 = x @ W1^T + b1
    {
        const int row_tiles = (B_ * LX) / 16;
        const int waves     = row_tiles * (DD / 16);   // 2048 waves
        dim3 grid((waves * 32 + 255) / 256), block(256);
        gemm16x16_wmma_f32<<<grid, block, 0, stream>>>(x, W1, b1, item1, row_tiles);
    }
    // item2 = memory @ W2^T
    {
        const int row_tiles = (B_ * LM) / 16;
        const int waves     = row_tiles * (DD / 16);
        dim3 grid((waves * 32 + 255) / 256), block(256);
        gemm16x16_wmma_f32<<<grid, block, 0, stream>>>(mem, W2, nullptr, item2, row_tiles);
    }
    // S = reduce_d tanh(item1 + item2) * Wt, masked
    {
        dim3 grid(LM / 32, LX / 8, B_), block(32, 8);
        tanh_attn_kernel<<<grid, block, 0, stream>>>(item1, item2, mask, Wt, out);
    }
}